// CARAFE_Downsample_85744727097492
// MI455X (gfx1250) — compile-verified
//
#include <hip/hip_runtime.h>
#include <hip/hip_bf16.h>

typedef __attribute__((ext_vector_type(2))) float v2f;
typedef __attribute__((ext_vector_type(8))) float v8f;

#define Bn   4
#define Cin  256
#define Him  128
#define Wim  128
#define COMP 64
#define KK   5
#define NK   25      // K*K
#define Ho   64
#define Wo   64
#define HW   (Him * Wim)     // 16384 pixels per image
#define HWo  (Ho * Wo)       // 4096 output pixels per image

// ---------------------------------------------------------------------------
// Kernel 1: compressed[b,m,h,w] = sum_c x[b,c,h,w] * w1[m,c] + b1[m]
// GEMM via V_WMMA_F32_16X16X4_F32. Block = 128 threads (4 waves).
// Each wave: M-tile of 16 channels (wave id), 4 N-tiles of 16 pixels.
// Block covers 64 channels x 64 pixels. Grid = 65536/64 = 1024 blocks.
// Weights staged in LDS as pairs so A-fragment is one ds_load_b64.
// ---------------------------------------------------------------------------
__global__ __launch_bounds__(128)
void carafe_conv1x1_wmma(const float* __restrict__ x,
                         const float* __restrict__ w1,
                         const float* __restrict__ b1,
                         float* __restrict__ comp) {
    // lds layout: index((c,m)) = ((c>>1)*COMP + m)*2 + (c&1)
    // -> the (c, c+1) pair a lane needs is contiguous (8B) for ds_load_b64
    __shared__ float lds_wp[Cin * COMP];   // 64 KB

    const int tid = threadIdx.x;
    // cooperative weight load, coalesced on global (c fastest in w1[m*Cin+c])
    for (int i = tid; i < Cin * COMP; i += 128) {
        int c = i & (Cin - 1);
        int m = i >> 8;
        lds_wp[(((c >> 1) * COMP) + m) * 2 + (c & 1)] = w1[m * Cin + c];
    }
    __syncthreads();

    const int lane = tid & 31;
    const int wv   = tid >> 5;          // 0..3 -> M-tile
    const int m0   = wv * 16;

    const int pblk  = blockIdx.x * 64;  // global pixel base (64 px per block)
    const int b     = pblk >> 14;       // image index (16384 px/image)
    const int p_img = pblk & (HW - 1);

    const int lh  = lane & 15;          // M (for A) / N (for B) within tile
    const int hi2 = (lane >> 4) << 1;   // +0 for lanes 0-15, +2 for 16-31

    const float* xb = x + (size_t)b * Cin * HW;

    v8f acc0 = {}, acc1 = {}, acc2 = {}, acc3 = {};

    for (int c0 = 0; c0 < Cin; c0 += 4) {
        // A fragment: lane holds W1[m0+lh][c0+hi2] , W1[m0+lh][c0+hi2+1]
        const int ca = c0 + hi2;                     // even
        v2f a = *(const v2f*)&lds_wp[(((ca >> 1) * COMP) + (m0 + lh)) * 2];

        // B fragments: lane holds X[ca][p] , X[ca+1][p]
        const float* xc0 = xb + (size_t)ca * HW + p_img + lh;
        const float* xc1 = xc0 + HW;
        v2f bf0, bf1, bf2, bf3;
        bf0.x = xc0[0];   bf0.y = xc1[0];
        bf1.x = xc0[16];  bf1.y = xc1[16];
        bf2.x = xc0[32];  bf2.y = xc1[32];
        bf3.x = xc0[48];  bf3.y = xc1[48];

        acc0 = __builtin_amdgcn_wmma_f32_16x16x4_f32(false, a, false, bf0,
                                                     (short)0, acc0, false, false);
        acc1 = __builtin_amdgcn_wmma_f32_16x16x4_f32(false, a, false, bf1,
                                                     (short)0, acc1, false, false);
        acc2 = __builtin_amdgcn_wmma_f32_16x16x4_f32(false, a, false, bf2,
                                                     (short)0, acc2, false, false);
        acc3 = __builtin_amdgcn_wmma_f32_16x16x4_f32(false, a, false, bf3,
                                                     (short)0, acc3, false, false);
    }

    // D layout: VGPR r, lanes 0-15 -> M=r, lanes 16-31 -> M=r+8; N = lane&15
    const int mhi = (lane >> 4) << 3;   // +8 for upper half-wave
#pragma unroll
    for (int r = 0; r < 8; ++r) {
        const int m    = m0 + r + mhi;
        const float bv = b1[m];
        float* op = comp + ((size_t)(b * COMP + m) * HW) + p_img + lh;
        op[0]  = acc0[r] + bv;
        op[16] = acc1[r] + bv;
        op[32] = acc2[r] + bv;
        op[48] = acc3[r] + bv;
    }
}

// ---------------------------------------------------------------------------
// Kernel 2: mask = softmax_k( conv3x3_s2_p1(compressed, w2) + b2 )
// One thread per (b,ho,wo). w2 (25*64*9 f32 = 57.6 KB) cached in LDS.
// ---------------------------------------------------------------------------
__global__ __launch_bounds__(256)
void carafe_mask_softmax(const float* __restrict__ comp,
                         const float* __restrict__ w2,
                         const float* __restrict__ b2,
                         float* __restrict__ mask) {
    __shared__ float lds_w2[NK * COMP * 9];   // [k][c][i][j], 57.6 KB

    const int tid = threadIdx.x;
    for (int i = tid; i < NK * COMP * 9; i += 256) lds_w2[i] = w2[i];
    __syncthreads();

    const int gid = blockIdx.x * 256 + tid;   // 0 .. 4*64*64-1
    const int wo  = gid & 63;
    const int ho  = (gid >> 6) & 63;
    const int b   = gid >> 12;

    float acc[NK];
#pragma unroll
    for (int k = 0; k < NK; ++k) acc[k] = b2[k];

    const float* cb = comp + (size_t)b * COMP * HW;
    for (int c = 0; c < COMP; ++c) {
        const float* cc = cb + (size_t)c * HW;
        for (int i = 0; i < 3; ++i) {
            const int hi = 2 * ho + i - 1;
            if ((unsigned)hi >= (unsigned)Him) continue;
            for (int j = 0; j < 3; ++j) {
                const int wi = 2 * wo + j - 1;
                if ((unsigned)wi >= (unsigned)Wim) continue;
                const float v = cc[hi * Wim + wi];
                const float* wrow = &lds_w2[c * 9 + i * 3 + j];
#pragma unroll
                for (int k = 0; k < NK; ++k)
                    acc[k] = fmaf(v, wrow[k * (COMP * 9)], acc[k]);
            }
        }
    }

    // softmax over 25
    float mx = acc[0];
#pragma unroll
    for (int k = 1; k < NK; ++k) mx = fmaxf(mx, acc[k]);
    float s = 0.f;
#pragma unroll
    for (int k = 0; k < NK; ++k) { acc[k] = __expf(acc[k] - mx); s += acc[k]; }
    const float inv = 1.0f / s;

    float* mb = mask + (size_t)b * NK * HWo + ho * Wo + wo;
#pragma unroll
    for (int k = 0; k < NK; ++k) mb[(size_t)k * HWo] = acc[k] * inv;
}

// ---------------------------------------------------------------------------
// Kernel 3: out[b,c,ho,wo] = sum_{i,j} x[b,c,2ho+i-2,2wo+j-2] * mask[b,ij,ho,wo]
// Block = 256 threads = (4 channels x 64 wo); mask row (25*64 f32) in LDS.
// Grid = b * ho * (256/4 channel-groups) = 4*64*64 = 16384 blocks.
// ---------------------------------------------------------------------------
__global__ __launch_bounds__(256)
void carafe_reassemble(const float* __restrict__ x,
                       const float* __restrict__ mask,
                       float* __restrict__ out) {
    __shared__ float m_lds[NK * Wo];   // 6.25 KB

    const int tid = threadIdx.x;
    const int bx  = blockIdx.x;
    const int cg  = bx & 63;            // channel group (4 channels each)
    const int ho  = (bx >> 6) & 63;
    const int b   = bx >> 12;

    // stage the 25 x 64 mask row for this (b, ho)
    const float* mb = mask + (size_t)b * NK * HWo + ho * Wo;
    for (int i = tid; i < NK * Wo; i += 256) {
        const int k  = i >> 6;
        const int wl = i & 63;
        m_lds[i] = mb[(size_t)k * HWo + wl];
    }
    __syncthreads();

    const int wo = tid & 63;
    const int c  = cg * 4 + (tid >> 6);

    const float* xb = x + (size_t)(b * Cin + c) * HW;
    float acc = 0.f;
#pragma unroll
    for (int i = 0; i < KK; ++i) {
        const int xi = 2 * ho + i - 2;
        if ((unsigned)xi >= (unsigned)Him) continue;
        const float* xr = xb + xi * Wim;
#pragma unroll
        for (int j = 0; j < KK; ++j) {
            const int xj = 2 * wo + j - 2;
            if ((unsigned)xj >= (unsigned)Wim) continue;
            acc = fmaf(xr[xj], m_lds[(i * KK + j) * Wo + wo], acc);
        }
    }
    out[((size_t)(b * Cin + c) * Ho + ho) * Wo + wo] = acc;
}

// ---------------------------------------------------------------------------
extern "C" void kernel_launch(void* const* d_in, const int* in_sizes, int n_in,
                              void* d_out, int out_size, void* d_ws, size_t ws_size,
                              hipStream_t stream) {
    const float* x  = (const float*)d_in[0];
    const float* w1 = (const float*)d_in[1];
    const float* b1 = (const float*)d_in[2];
    const float* w2 = (const float*)d_in[3];
    const float* b2 = (const float*)d_in[4];
    float* out = (float*)d_out;

    // workspace layout
    float* comp = (float*)d_ws;                                   // 16 MB
    float* mask = (float*)((char*)d_ws + (size_t)Bn * COMP * HW * sizeof(float));

    // Stage 1: conv1x1 GEMM via WMMA f32 — 1024 blocks x 128 threads
    carafe_conv1x1_wmma<<<(Bn * HW) / 64, 128, 0, stream>>>(x, w1, b1, comp);

    // Stage 2: mask conv + softmax — 4*64*64 threads
    carafe_mask_softmax<<<(Bn * HWo) / 256, 256, 0, stream>>>(comp, w2, b2, mask);

    // Stage 3: reassembly — 16384 blocks x 256 threads
    carafe_reassemble<<<Bn * 64 * 64, 256, 0, stream>>>(x, mask, out);
}